// ModelIntegrator_45140106281158
// MI455X (gfx1250) — compile-verified
//
#include <hip/hip_runtime.h>
#include <cstdint>
#include <cstddef>

#define NTIME 256
#define NHIST 6
#define NSIZE 8
#define NEWTON_ITERS 8
#define BLOCK 64

typedef __attribute__((ext_vector_type(4))) float v4f;   // native clang vector

// ---- CDNA5 async global->LDS staging (ASYNCcnt pipe) ----------------------
// GVS form: each lane copies 4 bytes from global saddr plus the u32 byte offset in vaddr to the LDS address in vdst
__device__ __forceinline__ void async_f32_to_lds(uint32_t lds_byte_addr,
                                                 const float* sbase,
                                                 uint32_t byte_off) {
    asm volatile("global_load_async_to_lds_b32 %0, %1, %2"
                 :
                 : "v"(lds_byte_addr), "v"(byte_off), "s"(sbase)
                 : "memory");
}
template <int N>
__device__ __forceinline__ void wait_async() {
    // async loads complete in order: waiting for <=N outstanding guarantees
    // all but the newest N have landed in LDS.
    asm volatile("s_wait_asynccnt %0" :: "i"(N) : "memory");
}
__device__ __forceinline__ uint32_t lds_addr_of(const void* p) {
    // low 32 bits of the flat LDS-aperture address are the LDS byte address
    return (uint32_t)(uintptr_t)p;
}

__global__ __launch_bounds__(BLOCK)
void visco_integrate_kernel(const float* __restrict__ times,
                            const float* __restrict__ strains,
                            const float* __restrict__ temps,
                            const float* __restrict__ pE,
                            const float* __restrict__ pn,
                            const float* __restrict__ peta,
                            const float* __restrict__ ps0,
                            const float* __restrict__ Cg,
                            const float* __restrict__ gg,
                            const float* __restrict__ pdcoef,
                            float* __restrict__ out,
                            int nbatch) {
    __shared__ float sT[2][BLOCK];   // times[t][b]  staging
    __shared__ float sS[2][BLOCK];   // strains[t][b]
    __shared__ float sQ[2][BLOCK];   // temperatures[t][b]

    const int tid = threadIdx.x;
    const int b   = blockIdx.x * BLOCK + tid;
    if (b >= nbatch) return;

    // uniform material constants
    const float E     = pE[0];
    const float n     = pn[0];
    const float eta   = peta[0];
    const float s0    = ps0[0];
    const float dcoef = pdcoef[0];
    const float inv_s0  = 1.0f / s0;
    const float inv_eta = 1.0f / eta;
    const float nm1     = n - 1.0f;

    float Cr[NHIST], gr[NHIST];
#pragma unroll
    for (int i = 0; i < NHIST; ++i) { Cr[i] = Cg[i]; gr[i] = gg[i]; }

    // state y = [s, h0..h5, d]; init zeros (D0 = 0)
    float s = 0.0f, dm = 0.0f, h[NHIST];
#pragma unroll
    for (int i = 0; i < NHIST; ++i) h[i] = 0.0f;

    // t = 0 output row: zeros (streaming, non-temporal b128 stores)
    {
        v4f z4 = (v4f)(0.0f);
        v4f* o = reinterpret_cast<v4f*>(out + (size_t)b * NSIZE);
        __builtin_nontemporal_store(z4, o);
        __builtin_nontemporal_store(z4, o + 1);
    }

    float prevT = times[b];     // times[0][b]
    float prevS = strains[b];   // strains[0][b]

    // prime the pipeline: async-stage t = 1 rows into buffer 0
    int cur = 0;
    {
        uint32_t off = (uint32_t)(((uint32_t)nbatch + (uint32_t)b) * 4u);
        async_f32_to_lds(lds_addr_of(&sT[0][tid]), times,   off);
        async_f32_to_lds(lds_addr_of(&sS[0][tid]), strains, off);
        async_f32_to_lds(lds_addr_of(&sQ[0][tid]), temps,   off);
    }

    for (int t = 1; t < NTIME; ++t) {
        // issue t+1 loads BEFORE consuming buffer t, then wait for the older
        // three only (in-order completion): keeps the async pipe one full
        // Newton solve ahead of consumption.
        if (t + 1 < NTIME) {
            int nxt = cur ^ 1;
            uint32_t off = (uint32_t)((((uint32_t)(t + 1)) * (uint32_t)nbatch
                                       + (uint32_t)b) * 4u);
            async_f32_to_lds(lds_addr_of(&sT[nxt][tid]), times,   off);
            async_f32_to_lds(lds_addr_of(&sS[nxt][tid]), strains, off);
            async_f32_to_lds(lds_addr_of(&sQ[nxt][tid]), temps,   off);
            wait_async<3>();
        } else {
            wait_async<0>();
        }
        float curT = sT[cur][tid];
        float curS = sS[cur][tid];
        float Temp = sQ[cur][tid];

        float dt = curT - prevT;
        float er = (curS - prevS) / dt;           // precise divide (once/step)
        // jnp.nan_to_num semantics
        if (er != er) er = 0.0f;
        else if (fabsf(er) > 3.4028234663852886e38f)
            er = copysignf(3.4028234663852886e38f, er);

        const float kv = __expf(-Temp * 0.001f) * inv_eta;  // hoisted per step

        // Newton iterate (starts at previous converged state)
        float sn = s, dn = dm, hn[NHIST];
#pragma unroll
        for (int i = 0; i < NHIST; ++i) hn[i] = h[i];

#pragma unroll
        for (int it = 0; it < NEWTON_ITERS; ++it) {
            float hsum = 0.0f;
#pragma unroll
            for (int i = 0; i < NHIST; ++i) hsum += hn[i];

            float over  = sn * (1.0f - dn) - hsum;
            float aover = fabsf(over);
            float sgn   = (over > 0.0f) ? 1.0f : ((over < 0.0f) ? -1.0f : 0.0f);
            float base  = aover * inv_s0;
            float pm1   = __powf(base, nm1);   // base^(n-1); base=0 -> 0
            float pw    = pm1 * base;          // base^n
            float afr   = pw * kv;             // |fr|
            float fr    = sgn * afr;
            float frp   = n * pm1 * inv_s0 * kv;  // d fr / d over >= 0

            // residual R = y_next - y - dt * f(y_next)   (rows 0 and 7)
            float R0 = sn - s  - dt * (E * (er - fr));
            float R7 = dn - dm - dt * (dcoef * afr);

            // A = I - dt*J = Dtilde - dt * p * w^T  (exact: J is diag+rank1)
            //   w = [(1-d), -1,...,-1, -s]
            //   Dtilde = diag(1, 1+dt*g_i*|fr|, 1)
            float u0 = -E * frp;
            float u7 = dcoef * sgn * frp;
            float z0 = R0, z7 = R7;            // diag entries are 1 there
            float wz = (1.0f - dn) * z0 - sn * z7;
            float wv = (1.0f - dn) * u0 - sn * u7;

            float zi[NHIST], vi[NHIST];
#pragma unroll
            for (int i = 0; i < NHIST; ++i) {
                float Ri = hn[i] - h[i]
                         - dt * (Cr[i] * fr - gr[i] * hn[i] * afr);
                // approximate-solve recip: only affects Newton convergence
                // path, not the fixed point R(y)=0.
                float di = __builtin_amdgcn_rcpf(1.0f + dt * gr[i] * afr);
                float ui = (Cr[i] - gr[i] * hn[i] * sgn) * frp;
                zi[i] = Ri * di;
                vi[i] = ui * di;
                wz -= zi[i];
                wv -= vi[i];
            }

            // Sherman-Morrison: x = z + dt*(w.z)/(1 - dt*(w.v)) * v
            float denom = 1.0f - dt * wv;
            float alpha = dt * wz * __builtin_amdgcn_rcpf(denom);

            sn -= z0 + alpha * u0;
            dn -= z7 + alpha * u7;
#pragma unroll
            for (int i = 0; i < NHIST; ++i) hn[i] -= zi[i] + alpha * vi[i];
        }

        s = sn; dm = dn;
#pragma unroll
        for (int i = 0; i < NHIST; ++i) h[i] = hn[i];

        // store row t: 8 floats = two non-temporal b128 stores
        v4f* o = reinterpret_cast<v4f*>(
            out + ((size_t)t * (size_t)nbatch + (size_t)b) * NSIZE);
        v4f lo, hi;
        lo.x = s;    lo.y = h[0]; lo.z = h[1]; lo.w = h[2];
        hi.x = h[3]; hi.y = h[4]; hi.z = h[5]; hi.w = dm;
        __builtin_nontemporal_store(lo, o);
        __builtin_nontemporal_store(hi, o + 1);

        prevT = curT;
        prevS = curS;
        cur ^= 1;
    }
}

extern "C" void kernel_launch(void* const* d_in, const int* in_sizes, int n_in,
                              void* d_out, int out_size, void* d_ws, size_t ws_size,
                              hipStream_t stream) {
    const float* times   = (const float*)d_in[0];
    const float* strains = (const float*)d_in[1];
    const float* temps   = (const float*)d_in[2];
    const float* E       = (const float*)d_in[3];
    const float* n       = (const float*)d_in[4];
    const float* eta     = (const float*)d_in[5];
    const float* s0      = (const float*)d_in[6];
    const float* C       = (const float*)d_in[7];
    const float* g       = (const float*)d_in[8];
    const float* dcoef   = (const float*)d_in[9];
    float* out = (float*)d_out;

    int nbatch = in_sizes[0] / NTIME;
    int grid   = (nbatch + BLOCK - 1) / BLOCK;   // 128 blocks of 64 threads
    visco_integrate_kernel<<<grid, BLOCK, 0, stream>>>(
        times, strains, temps, E, n, eta, s0, C, g, dcoef, out, nbatch);
}